// Universal_model_54056458387462
// MI455X (gfx1250) — compile-verified
//
#include <hip/hip_runtime.h>

typedef __attribute__((ext_vector_type(2))) float v2f;
typedef __attribute__((ext_vector_type(8))) float v8f;

#define HWF 65536   // 256*256

__device__ __forceinline__ float lane_shfl(float v, int srcLane) {
  return __int_as_float(__builtin_amdgcn_ds_bpermute(srcLane << 2, __float_as_int(v)));
}

// ---------------- GroupNorm statistics: one block per (b, group) -------------
__global__ void gn_stats_kernel(const float* __restrict__ x, float* __restrict__ stats,
                                int elems_per_group, float inv_n) {
  __shared__ float s_sum[256], s_sq[256];
  const float* p = x + (size_t)blockIdx.x * (size_t)elems_per_group;
  float s = 0.f, ss = 0.f;
  for (int i = threadIdx.x; i < elems_per_group; i += 256) {
    float v = p[i]; s += v; ss = fmaf(v, v, ss);
  }
  s_sum[threadIdx.x] = s; s_sq[threadIdx.x] = ss;
  __syncthreads();
  for (int off = 128; off > 0; off >>= 1) {
    if (threadIdx.x < off) {
      s_sum[threadIdx.x] += s_sum[threadIdx.x + off];
      s_sq[threadIdx.x]  += s_sq[threadIdx.x + off];
    }
    __syncthreads();
  }
  if (threadIdx.x == 0) {
    float mu  = s_sum[0] * inv_n;
    float var = s_sq[0] * inv_n - mu * mu;
    stats[2 * blockIdx.x]     = mu;
    stats[2 * blockIdx.x + 1] = rsqrtf(var + 1e-5f);
  }
}

// ------- fused GN + ReLU + 1x1 conv (W_pre 8x64): feat -> head_in (B,8,HW) ---
__global__ void head_in_kernel(const float* __restrict__ feat,
                               const float* __restrict__ stats,
                               const float* __restrict__ gamma,
                               const float* __restrict__ beta,
                               const float* __restrict__ Wpre,
                               const float* __restrict__ bpre,
                               float* __restrict__ head) {
  __shared__ float sW[512];          // transposed: [c][o], o contiguous
  __shared__ float sA[64], sD[64];   // per-channel scale/shift
  const int tid = threadIdx.x;
  const int b = blockIdx.y;
  const int pix = blockIdx.x * 256 + tid;
  {
    int o = tid >> 6, c = tid & 63;
    sW[c * 8 + o] = Wpre[tid];                 // tid = o*64+c
    o = (tid + 256) >> 6; c = tid & 63;
    sW[c * 8 + o] = Wpre[tid + 256];
  }
  if (tid < 64) {
    int c = tid, g = c >> 2;
    float mu = stats[2 * (b * 16 + g)];
    float rs = stats[2 * (b * 16 + g) + 1];
    float a = rs * gamma[c];
    sA[c] = a;
    sD[c] = beta[c] - mu * a;
  }
  __syncthreads();

  float acc[8];
#pragma unroll
  for (int o = 0; o < 8; ++o) acc[o] = bpre[o];

  const float* fp = feat + (size_t)b * 64 * HWF + pix;
#pragma unroll 8
  for (int c = 0; c < 64; ++c) {
    float v = fp[(size_t)c * HWF];
    v = fmaxf(fmaf(v, sA[c], sD[c]), 0.f);
#pragma unroll
    for (int o = 0; o < 8; ++o) acc[o] = fmaf(sW[c * 8 + o], v, acc[o]);
  }
  float* hp = head + (size_t)b * 8 * HWF + pix;
#pragma unroll
  for (int o = 0; o < 8; ++o) hp[(size_t)o * HWF] = acc[o];
}

// ----------- dec4: GN + ReLU + spatial mean -> g (B,512); 1 block per (b,ch) -
__global__ void gap_kernel(const float* __restrict__ dec4, const float* __restrict__ stats,
                           const float* __restrict__ gamma, const float* __restrict__ beta,
                           float* __restrict__ g) {
  __shared__ float red[256];
  const int idx = blockIdx.x;              // b*512 + ch
  const int b = idx >> 9, ch = idx & 511;
  const int grp = ch >> 5;
  const float mu = stats[2 * (b * 16 + grp)];
  const float rs = stats[2 * (b * 16 + grp) + 1];
  float v = dec4[(size_t)idx * 256 + threadIdx.x];
  v = fmaxf((v - mu) * rs * gamma[ch] + beta[ch], 0.f);
  red[threadIdx.x] = v;
  __syncthreads();
  for (int off = 128; off > 0; off >>= 1) {
    if (threadIdx.x < off) red[threadIdx.x] += red[threadIdx.x + off];
    __syncthreads();
  }
  if (threadIdx.x == 0) g[idx] = red[0] * (1.f / 256.f);
}

// ----------------- x_feat = g @ W_gap^T + b_gap : (B,256) --------------------
__global__ void xfeat_kernel(const float* __restrict__ g, const float* __restrict__ Wgap,
                             const float* __restrict__ bgap, float* __restrict__ xfeat) {
  __shared__ float sg[512];
  const int b = blockIdx.x;
  sg[threadIdx.x]       = g[b * 512 + threadIdx.x];
  sg[threadIdx.x + 256] = g[b * 512 + 256 + threadIdx.x];
  __syncthreads();
  const int o = threadIdx.x;
  const float* w = Wgap + (size_t)o * 512;
  float acc = bgap[o];
#pragma unroll 8
  for (int i = 0; i < 512; ++i) acc = fmaf(w[i], sg[i], acc);
  xfeat[b * 256 + o] = acc;
}

// -------- params = [x_feat | emb] @ W_ctrl^T + b_ctrl : (B*C,153) ------------
__global__ void params_kernel(const float* __restrict__ xfeat, const float* __restrict__ emb,
                              const float* __restrict__ Wctrl, const float* __restrict__ bctrl,
                              float* __restrict__ params) {
  __shared__ float sx[512];
  const int b = blockIdx.x >> 5;           // blockIdx.x = b*32 + c
  const int c = blockIdx.x & 31;
  sx[threadIdx.x]       = xfeat[b * 256 + threadIdx.x];
  sx[256 + threadIdx.x] = emb[c * 256 + threadIdx.x];
  __syncthreads();
  const int o = threadIdx.x;
  if (o < 153) {
    const float* w = Wctrl + (size_t)o * 512;
    float acc = bctrl[o];
#pragma unroll 8
    for (int i = 0; i < 512; ++i) acc = fmaf(w[i], sx[i], acc);
    params[(size_t)blockIdx.x * 153 + o] = acc;
  }
}

// ---------------- dynamic per-(b,c) 8->8->8->1 MLP over pixels ---------------
// Wave = one (b, c-pair); tile = 16 pixels. Layer1: 2x WMMA f32 16x16x4 with
// A=[w1_c0;w1_c1]. Layer2: block-diag diag(w2_c0,w2_c1), 4x WMMA after a
// ds_bpermute relayout of x1 (D-layout -> B-layout). Layer3: lane-local dot.
__global__ void mlp_head_kernel(const float* __restrict__ head,
                                const float* __restrict__ params,
                                float* __restrict__ out) {
  const int tid  = threadIdx.x;
  const int L    = tid & 31;
  const int wave = tid >> 5;
  const int wgid = blockIdx.x * 8 + wave;   // 0..16383
  const int cpair  = wgid & 63;             // b*16 + cp
  const int tstart = wgid >> 6;             // 0..255
  const int b  = cpair >> 4;
  const int c0 = (cpair & 15) * 2;
  const int c1 = c0 + 1;

  const float* p0 = params + (size_t)(b * 32 + c0) * 153;
  const float* p1 = params + (size_t)(b * 32 + c1) * 153;

  const int  n     = L & 15;               // column (pixel within tile)
  const bool hi    = (L >= 16);
  const int  halfk = hi ? 2 : 0;
  const int  M     = L & 15;               // A-matrix row held by this lane
  const int  Mr    = M & 7;
  const float* pwA = (M < 8) ? p0 : p1;    // layer-1 A row source

  // Layer-1 A operands: combined w1 (16x8), K split into 2 x (K=4)
  v2f a1[2];
#pragma unroll
  for (int j = 0; j < 2; ++j) {
    a1[j].x = pwA[Mr * 8 + 4 * j + halfk];
    a1[j].y = pwA[Mr * 8 + 4 * j + halfk + 1];
  }
  // Layer-2 A operands: block-diag [[w2_c0,0],[0,w2_c1]] (16x16), 4 x (K=4)
  v2f a2[4];
#pragma unroll
  for (int j = 0; j < 4; ++j) {
#pragma unroll
    for (int v = 0; v < 2; ++v) {
      const int k = 4 * j + halfk + v;
      float wlo = p0[64 + Mr * 8 + (k & 7)];
      float whi = p1[64 + Mr * 8 + (k & 7)];
      const bool topM = (M >= 8), topK = (k >= 8);
      float val = topM ? (topK ? whi : 0.f) : (topK ? 0.f : wlo);
      if (v == 0) a2[j].x = val; else a2[j].y = val;
    }
  }
  // Per-lane (lane-half selects c0 vs c1) biases + final weights
  const float* pb = hi ? p1 : p0;
  float b1s[8], b2s[8], w3s[8];
#pragma unroll
  for (int k = 0; k < 8; ++k) {
    w3s[k] = pb[128 + k];
    b1s[k] = pb[136 + k];
    b2s[k] = pb[144 + k];
  }
  const float b3s = pb[152];

  const float* hbase = head + (size_t)b * 8 * HWF;
  const int cOut = hi ? c1 : c0;
  float* obase = out + (size_t)(b * 32 + cOut) * HWF + n;

  for (int t = 0; t < 16; ++t) {
    const int pixbase = (tstart + t * 256) * 16;

    // Layer-1 B operands: head_in rows (K) x 16 pixel columns
    v2f bop[2];
#pragma unroll
    for (int j = 0; j < 2; ++j) {
      const int k = 4 * j + halfk;
      bop[j].x = hbase[(size_t)k * HWF + pixbase + n];
      bop[j].y = hbase[(size_t)(k + 1) * HWF + pixbase + n];
    }

    v8f acc = {};
    acc = __builtin_amdgcn_wmma_f32_16x16x4_f32(false, a1[0], false, bop[0], (short)0, acc, false, false);
    acc = __builtin_amdgcn_wmma_f32_16x16x4_f32(false, a1[1], false, bop[1], (short)0, acc, false, false);

    // x1 per lane: D-layout => lane holds full 8-vector of its (c, pixel)
    float x1v[8];
#pragma unroll
    for (int r = 0; r < 8; ++r) x1v[r] = fmaxf(acc[r] + b1s[r], 0.f);

    // Relayout x1 (D-layout) -> combined 16x16 B operands for layer 2
    v2f b2op[4];
#pragma unroll
    for (int j = 0; j < 4; ++j) {
#pragma unroll
      for (int v = 0; v < 2; ++v) {
        const int rl = 4 * j + v;      // row for lanes 0-15
        const int rh = rl + 2;         // row for lanes 16-31
        float tlo = lane_shfl(x1v[rl & 7], n + ((rl >= 8) ? 16 : 0));
        float thi = lane_shfl(x1v[rh & 7], n + ((rh >= 8) ? 16 : 0));
        float val = hi ? thi : tlo;
        if (v == 0) b2op[j].x = val; else b2op[j].y = val;
      }
    }

    v8f acc2 = {};
    acc2 = __builtin_amdgcn_wmma_f32_16x16x4_f32(false, a2[0], false, b2op[0], (short)0, acc2, false, false);
    acc2 = __builtin_amdgcn_wmma_f32_16x16x4_f32(false, a2[1], false, b2op[1], (short)0, acc2, false, false);
    acc2 = __builtin_amdgcn_wmma_f32_16x16x4_f32(false, a2[2], false, b2op[2], (short)0, acc2, false, false);
    acc2 = __builtin_amdgcn_wmma_f32_16x16x4_f32(false, a2[3], false, b2op[3], (short)0, acc2, false, false);

    // Layer 3: lane-local dot + bias, ReLU applied to x2
    float logit = b3s;
#pragma unroll
    for (int r = 0; r < 8; ++r) {
      float x2 = fmaxf(acc2[r] + b2s[r], 0.f);
      logit = fmaf(w3s[r], x2, logit);
    }
    obase[pixbase] = logit;
  }
}

extern "C" void kernel_launch(void* const* d_in, const int* in_sizes, int n_in,
                              void* d_out, int out_size, void* d_ws, size_t ws_size,
                              hipStream_t stream) {
  const float* dec4   = (const float*)d_in[0];
  const float* feat   = (const float*)d_in[1];
  const float* gpre_g = (const float*)d_in[2];
  const float* gpre_b = (const float*)d_in[3];
  const float* Wpre   = (const float*)d_in[4];
  const float* bpre   = (const float*)d_in[5];
  const float* ggap_g = (const float*)d_in[6];
  const float* ggap_b = (const float*)d_in[7];
  const float* Wgap   = (const float*)d_in[8];
  const float* bgap   = (const float*)d_in[9];
  const float* emb    = (const float*)d_in[10];
  const float* Wctrl  = (const float*)d_in[11];
  const float* bctrl  = (const float*)d_in[12];
  float* out = (float*)d_out;

  char* ws = (char*)d_ws;
  float* statsF = (float*)(ws);                 // 128 f
  float* statsD = (float*)(ws + 1024);          // 128 f
  float* gbuf   = (float*)(ws + 2048);          // 2048 f
  float* xfeat  = (float*)(ws + 2048 + 8192);   // 1024 f
  float* parb   = (float*)(ws + 16384);         // 19584 f
  float* head   = (float*)(ws + 131072);        // 8 MB (B*8*HW f)

  gn_stats_kernel<<<64, 256, 0, stream>>>(feat, statsF, 4 * HWF, 1.f / (4.f * HWF));
  gn_stats_kernel<<<64, 256, 0, stream>>>(dec4, statsD, 32 * 256, 1.f / (32.f * 256.f));
  head_in_kernel<<<dim3(HWF / 256, 4), 256, 0, stream>>>(feat, statsF, gpre_g, gpre_b, Wpre, bpre, head);
  gap_kernel<<<4 * 512, 256, 0, stream>>>(dec4, statsD, ggap_g, ggap_b, gbuf);
  xfeat_kernel<<<4, 256, 0, stream>>>(gbuf, Wgap, bgap, xfeat);
  params_kernel<<<4 * 32, 256, 0, stream>>>(xfeat, emb, Wctrl, bctrl, parb);
  mlp_head_kernel<<<2048, 256, 0, stream>>>(head, parb, out);
}